// DynamicProjections_25451976196906
// MI455X (gfx1250) — compile-verified
//
#include <hip/hip_runtime.h>

typedef float v2f __attribute__((ext_vector_type(2)));
typedef float v4f __attribute__((ext_vector_type(4)));
typedef float v8f __attribute__((ext_vector_type(8)));

#define B_    8
#define S_    2048
#define DIM_  1024
#define H_    256
#define C_    32
#define NCHUNK (S_ / C_)
#define ETA_  0.01f

// ---- WMMA fragment helpers (fp32, 16x16x4; layouts per CDNA5 ISA 7.12.2) ----

// A 16x4 (MxK), or B 4x16 sourced as the transpose of a row-major matrix:
// lane l<16: {row r0+l, cols c0,c0+1}; lane l>=16: {row r0+l-16, cols c0+2,c0+3}
__device__ __forceinline__ v2f frag_rowpair(const float* base, int stride, int r0, int c0, int lane) {
  int row = r0 + (lane & 15);
  int col = c0 + ((lane >> 4) << 1);
  const float* p = base + (size_t)row * (size_t)stride + col;
  v2f r; r.x = p[0]; r.y = p[1]; return r;
}

// B 4x16 (KxN) read directly from a row-major [K][N] source:
// lane l: N = c0+(l&15); v0 = row r0+2*(l>>4), v1 = row +1
__device__ __forceinline__ v2f frag_colpair(const float* base, int stride, int r0, int c0, int lane) {
  int col = c0 + (lane & 15);
  int row = r0 + ((lane >> 4) << 1);
  v2f r;
  r.x = base[(size_t)row * (size_t)stride + col];
  r.y = base[(size_t)(row + 1) * (size_t)stride + col];
  return r;
}

// C/D 16x16 f32: VGPR r -> M = r0 + r + 8*(lane>=16), N = c0 + (lane&15)
__device__ __forceinline__ v8f load_cfrag(const float* base, int stride, int r0, int c0, int lane) {
  int col = c0 + (lane & 15);
  int rb  = r0 + ((lane >> 4) << 3);
  v8f c;
#pragma unroll
  for (int r = 0; r < 8; ++r) c[r] = base[(size_t)(rb + r) * (size_t)stride + col];
  return c;
}

__device__ __forceinline__ void store_cfrag(float* base, int stride, int r0, int c0, int lane, v8f c) {
  int col = c0 + (lane & 15);
  int rb  = r0 + ((lane >> 4) << 3);
#pragma unroll
  for (int r = 0; r < 8; ++r) base[(size_t)(rb + r) * (size_t)stride + col] = c[r];
}

#define WMMA_ACC(a, b, c) \
  __builtin_amdgcn_wmma_f32_16x16x4_f32(false, (a), false, (b), (short)0, (c), false, false)

#define V8F_ZERO {0.f, 0.f, 0.f, 0.f, 0.f, 0.f, 0.f, 0.f}

// ---- CDNA5 async global -> LDS copy (ASYNCcnt-tracked), per CDNA5 ISA ----
__device__ __forceinline__ void async_copy_b128(const float* lds_dst, const float* gsrc) {
  uint32_t lds = (uint32_t)(uintptr_t)lds_dst;            // flat addr[31:0] == LDS offset
  uint64_t ga  = (uint64_t)(uintptr_t)gsrc;
  asm volatile("global_load_async_to_lds_b128 %0, %1, off"
               :: "v"(lds), "v"(ga) : "memory");
}
__device__ __forceinline__ void wait_async_zero() {
  asm volatile("s_wait_asynccnt 0x0" ::: "memory");
}

// Issue one chunk's X (32x1024 f32) + target slice (32x64 f32) into LDS buffers.
__device__ __forceinline__ void issue_chunk_async(const float* xb, const float* tb,
                                                  int t_base, int h_base, int tid,
                                                  float* Xbuf, float* Tbuf) {
  const float* gx = xb + (size_t)t_base * DIM_;
  for (int i = tid; i < C_ * DIM_ / 4; i += 64)
    async_copy_b128(Xbuf + i * 4, gx + i * 4);
  const float* gt = tb + (size_t)t_base * H_ + h_base;
  for (int i = tid; i < C_ * 64 / 4; i += 64) {
    int t = i >> 4, j = i & 15;
    async_copy_b128(Tbuf + t * 64 + j * 4, gt + (size_t)t * H_ + j * 4);
  }
}

// One workgroup = (batch b, matrix m, 64-row group g). 64 threads = 2 wave32.
// Wave w owns local rows [w*32, w*32+32): all global W traffic is wave-private.
__global__ __launch_bounds__(64)
void deltanet_scan_kernel(const float* __restrict__ x,
                          const float* __restrict__ tgt,
                          const float* __restrict__ Wq0,
                          const float* __restrict__ Wk0,
                          const float* __restrict__ Wv0,
                          float* __restrict__ out) {
  __shared__ float Xs[2][C_ * DIM_];  // 256 KB: double-buffered x chunk [t][d]
  __shared__ float Ts[2][C_ * 64];    //  16 KB: double-buffered targets [t][h_local]
  __shared__ float Ps[64 * C_];       //   8 KB: P = W0 X^T [h_local][t]
  __shared__ float Us[64 * C_];       //   8 KB: -U [h_local][t]
  __shared__ float Gs[C_ * C_];       //   4 KB: G = X X^T [s][t]

  const int tid  = threadIdx.x;
  const int lane = tid & 31;
  const int wv   = tid >> 5;

  const int gid    = blockIdx.x;  // [0, 96)
  const int bm     = gid >> 2;    // [0, 24)
  const int g      = gid & 3;
  const int m      = bm >> 3;     // matrix 0..2
  const int b      = bm & 7;      // batch
  const int h_base = g * 64;

  const size_t BSH = (size_t)B_ * S_ * H_;
  const size_t HD  = (size_t)H_ * DIM_;

  const float* W0   = (m == 0) ? Wq0 : ((m == 1) ? Wk0 : Wv0);
  float*       Wg   = out + 3 * BSH + ((size_t)m * B_ + b) * HD;  // fast weights (== final output)
  float*       outM = out + (size_t)m * BSH + (size_t)b * S_ * H_;
  const float* xb   = x   + (size_t)b * S_ * DIM_;
  const float* tb   = tgt + (size_t)b * S_ * H_;

  // Kick off chunk 0 staging while we init the fast weights.
  issue_chunk_async(xb, tb, 0, h_base, tid, Xs[0], Ts[0]);

  // Init this WG's 64 rows of fast weights; rows are wave-private.
  {
    const v4f* src = (const v4f*)(W0 + (size_t)(h_base + wv * 32) * DIM_);
    v4f*       dst = (v4f*)(Wg + (size_t)(h_base + wv * 32) * DIM_);
    for (int i = lane; i < 32 * DIM_ / 4; i += 32) dst[i] = src[i];
  }
  wait_async_zero();
  __syncthreads();

  for (int c0 = 0; c0 < NCHUNK; ++c0) {
    const int t_base = c0 * C_;
    const int cur    = c0 & 1;
    const float* Xc  = Xs[cur];

    // Overlap next chunk's staging with this chunk's compute.
    if (c0 + 1 < NCHUNK)
      issue_chunk_async(xb, tb, t_base + C_, h_base, tid, Xs[cur ^ 1], Ts[cur ^ 1]);

    // ---- fused  G = X X^T  and  P = W X^T : 6 independent WMMA chains/wave ----
    {
      const int r0 = wv * 32;  // this wave's local row strip base
      v8f g0 = V8F_ZERO, g1 = V8F_ZERO;
      v8f p00 = V8F_ZERO, p01 = V8F_ZERO, p10 = V8F_ZERO, p11 = V8F_ZERO;
      for (int d0 = 0; d0 < DIM_; d0 += 4) {
        v2f b0 = frag_rowpair(Xc, DIM_, 0, d0, lane);    // B[k,n] = X[n][d0+k]
        v2f b1 = frag_rowpair(Xc, DIM_, 16, d0, lane);
        v2f ag = frag_rowpair(Xc, DIM_, wv * 16, d0, lane);
        v2f a0 = frag_rowpair(Wg, DIM_, h_base + r0, d0, lane);       // L2-resident
        v2f a1 = frag_rowpair(Wg, DIM_, h_base + r0 + 16, d0, lane);
        g0  = WMMA_ACC(ag, b0, g0);
        g1  = WMMA_ACC(ag, b1, g1);
        p00 = WMMA_ACC(a0, b0, p00);
        p01 = WMMA_ACC(a0, b1, p01);
        p10 = WMMA_ACC(a1, b0, p10);
        p11 = WMMA_ACC(a1, b1, p11);
      }
      store_cfrag(Gs, C_, wv * 16, 0, lane, g0);
      store_cfrag(Gs, C_, wv * 16, 16, lane, g1);
      store_cfrag(Ps, C_, r0, 0, lane, p00);
      store_cfrag(Ps, C_, r0, 16, lane, p01);
      store_cfrag(Ps, C_, r0 + 16, 0, lane, p10);
      store_cfrag(Ps, C_, r0 + 16, 16, lane, p11);
    }
    __syncthreads();

    // ---- within-chunk sequential recurrence: one head-row per thread ----
    {
      float a[C_];
#pragma unroll
      for (int t = 0; t < C_; ++t) a[t] = Ps[tid * C_ + t];
#pragma unroll
      for (int t = 0; t < C_; ++t) {
        const float pred = a[t];  // W_t x_t
        const float u = (1.0f + ETA_) * pred - ETA_ * Ts[cur][t * 64 + tid];
        Us[tid * C_ + t] = -u;    // store -u so step D is pure multiply-add
        outM[(size_t)(t_base + t) * H_ + (h_base + tid)] = pred;  // q/k/v output
#pragma unroll
        for (int s = t + 1; s < C_; ++s) a[s] = fmaf(-Gs[t * C_ + s], u, a[s]);
      }
    }
    __syncthreads();

    // ---- W += (-U) X : 2 strips per wave, 64 N-tiles, K=32 ----
    for (int st = 0; st < 2; ++st) {
      const int r0 = wv * 32 + st * 16;
      v2f af[8];
#pragma unroll
      for (int kk = 0; kk < 8; ++kk) af[kk] = frag_rowpair(Us, C_, r0, kk * 4, lane);
#pragma unroll 2
      for (int n0 = 0; n0 < DIM_; n0 += 16) {
        v8f cc = load_cfrag(Wg, DIM_, h_base + r0, n0, lane);
#pragma unroll
        for (int kk = 0; kk < 8; ++kk) {
          v2f bf = frag_colpair(Xc, DIM_, kk * 4, n0, lane);  // B[k,n] = X[k][n0+n]
          cc = WMMA_ACC(af[kk], bf, cc);
        }
        store_cfrag(Wg, DIM_, h_base + r0, n0, lane, cc);
      }
    }

    // Drain this wave's async copies, then make the next buffer visible WG-wide.
    wait_async_zero();
    __syncthreads();
  }
}

extern "C" void kernel_launch(void* const* d_in, const int* in_sizes, int n_in,
                              void* d_out, int out_size, void* d_ws, size_t ws_size,
                              hipStream_t stream) {
  (void)in_sizes; (void)n_in; (void)out_size; (void)d_ws; (void)ws_size;
  const float* x   = (const float*)d_in[0];
  const float* tgt = (const float*)d_in[1];
  const float* Wq0 = (const float*)d_in[2];
  const float* Wk0 = (const float*)d_in[3];
  const float* Wv0 = (const float*)d_in[4];
  float* out = (float*)d_out;

  dim3 grid(96);   // 8 batches * 3 matrices * 4 row-groups
  dim3 block(64);  // 2 wave32
  hipLaunchKernelGGL(deltanet_scan_kernel, grid, block, 0, stream,
                     x, tgt, Wq0, Wk0, Wv0, out);
}